// AmplitudeModel_55963423867441
// MI455X (gfx1250) — compile-verified
//
#include <hip/hip_runtime.h>
#include <math.h>

// CDNA5 / gfx1250: wave32, f32 WMMA 16x16x4
typedef __attribute__((ext_vector_type(2))) float v2f;
typedef __attribute__((ext_vector_type(8))) float v8f;

#define NQ      12
#define DIM     4096        // 2^12
#define THREADS 256         // 8 wave32 waves

// One workgroup per batch row. Full complex state (32KB) lives in LDS,
// processed with 4 fused 3-qubit stages, each a 16x16 real-embedded gate
// matrix applied via chained V_WMMA_F32_16X16X4_F32.
__global__ __launch_bounds__(THREADS)
void amp_model_kernel(const float* __restrict__ inputs,
                      const float* __restrict__ observable,
                      const float* __restrict__ w,
                      float* __restrict__ out)
{
    __shared__ __align__(16) float reA[DIM];
    __shared__ __align__(16) float imA[DIM];
    __shared__ __align__(16) float reB[DIM];
    __shared__ __align__(16) float imB[DIM];
    __shared__ float Wr[4][16][16];   // real-embedded fused gate per stage
    __shared__ float Ur[NQ][4];       // 2x2 complex rotation matrices
    __shared__ float Ui[NQ][4];
    __shared__ float Osh[16];         // 4x4 observable
    __shared__ float red[THREADS];

    const int tid  = threadIdx.x;
    const int b    = blockIdx.x;
    const int lane = tid & 31;
    const int wv   = tid >> 5;
    const int half = lane >> 4;   // 0: lanes 0-15, 1: lanes 16-31
    const int row  = lane & 15;

    // ---- load input row into LDS, accumulate sum of squares ----
    const float* x = inputs + (size_t)b * DIM;
    float ss = 0.f;
    #pragma unroll
    for (int i = tid; i < DIM/4; i += THREADS) {
        float4 v = ((const float4*)x)[i];
        ((float4*)reA)[i] = v;
        ss += v.x*v.x + v.y*v.y + v.z*v.z + v.w*v.w;
    }

    // ---- per-qubit rotation matrices from w (12 x 3) ----
    if (tid < NQ) {
        float ph = w[tid*3+0], th = w[tid*3+1], om = w[tid*3+2];
        float c  = cosf(0.5f*th), s = sinf(0.5f*th);
        float a  = 0.5f*(ph+om),  bb = 0.5f*(ph-om);
        float ca = cosf(a),  sa = sinf(a);
        float cb = cosf(bb), sb = sinf(bb);
        Ur[tid][0] =  ca*c;  Ui[tid][0] = -sa*c;   // ep*c
        Ur[tid][1] = -cb*s;  Ui[tid][1] = -sb*s;   // -conj(em)*s
        Ur[tid][2] =  cb*s;  Ui[tid][2] = -sb*s;   // em*s
        Ur[tid][3] =  ca*c;  Ui[tid][3] =  sa*c;   // conj(ep)*c
    }
    if (tid < 16) Osh[tid] = observable[tid];

    red[tid] = ss;
    __syncthreads();
    for (int off = THREADS/2; off > 0; off >>= 1) {
        if (tid < off) red[tid] += red[tid + off];
        __syncthreads();
    }
    const float invn = 1.0f / sqrtf(red[0]);
    __syncthreads();

    // ---- normalize state, zero imaginary part ----
    #pragma unroll
    for (int i = tid; i < DIM; i += THREADS) {
        reA[i] *= invn;
        imA[i]  = 0.f;
    }

    // ---- build 4 fused 3-qubit gates: Wg = U[3g] (x) U[3g+1] (x) U[3g+2],
    //      real-embedded as 16x16: [[Re,-Im],[Im,Re]]. One entry/thread/stage.
    {
        const int I = tid >> 4, J = tid & 15;
        const int I8 = I & 7,  J8 = J & 7;
        const int i0 = (I8>>2)&1, i1 = (I8>>1)&1, i2 = I8&1;
        const int j0 = (J8>>2)&1, j1 = (J8>>1)&1, j2 = J8&1;
        #pragma unroll
        for (int g = 0; g < 4; ++g) {
            float ar = Ur[3*g  ][i0*2+j0], ai = Ui[3*g  ][i0*2+j0];
            float br = Ur[3*g+1][i1*2+j1], bi = Ui[3*g+1][i1*2+j1];
            float cr = Ur[3*g+2][i2*2+j2], ci = Ui[3*g+2][i2*2+j2];
            float pr = ar*br - ai*bi,  pi = ar*bi + ai*br;
            float wr = pr*cr - pi*ci,  wi = pr*ci + pi*cr;
            float val = (I < 8) ? ((J < 8) ? wr : -wi)
                                : ((J < 8) ? wi :  wr);
            Wr[g][I][J] = val;
        }
    }
    __syncthreads();

    // ---- 4 WMMA stages, ping-pong LDS buffers ----
    float* srcRe = reA; float* srcIm = imA;
    float* dstRe = reB; float* dstIm = imB;

    #pragma unroll
    for (int g = 0; g < 4; ++g) {
        const int lb = 9 - 3*g;                      // bit position of J group
        const unsigned lowmask = (1u << lb) - 1u;

        // A fragments (gate matrix), constant across tiles of this stage.
        // A 16x4 layout: lane M = row, VGPR k-pair at K = 4k + 2*half.
        v2f a0, a1, a2, a3;
        {
            const float* wrow = &Wr[g][row][0];
            const int bse = 2*half;
            a0.x = wrow[bse+0];   a0.y = wrow[bse+1];
            a1.x = wrow[bse+4];   a1.y = wrow[bse+5];
            a2.x = wrow[bse+8];   a2.y = wrow[bse+9];
            a3.x = wrow[bse+12];  a3.y = wrow[bse+13];
        }

        // 32 column-tiles over 8 waves: exactly 4 tiles per wave.
        // Compile-time trip count -> no EXEC masking, full unroll & pipelining.
        #pragma unroll
        for (int it = 0; it < 4; ++it) {
            const int t   = wv + 8*it;
            const int col = t*16 + row;
            const unsigned mbase = (((unsigned)col >> lb) << (lb + 3))
                                 | ((unsigned)col & lowmask);
            const int J0 = 2*half;
            const unsigned m0 = mbase | ((unsigned)(J0    ) << lb);
            const unsigned m1 = mbase | ((unsigned)(J0 + 1) << lb);
            const unsigned m4 = mbase | ((unsigned)(J0 + 4) << lb);
            const unsigned m5 = mbase | ((unsigned)(J0 + 5) << lb);

            // B 4x16 fragments: rows of X = [Re(v); Im(v)] at this tile.
            v2f b0, b1, b2, b3;
            b0.x = srcRe[m0]; b0.y = srcRe[m1];
            b1.x = srcRe[m4]; b1.y = srcRe[m5];
            b2.x = srcIm[m0]; b2.y = srcIm[m1];
            b3.x = srcIm[m4]; b3.y = srcIm[m5];

            v8f d = {};
            d = __builtin_amdgcn_wmma_f32_16x16x4_f32(false, a0, false, b0, (short)0, d, false, false);
            d = __builtin_amdgcn_wmma_f32_16x16x4_f32(false, a1, false, b1, (short)0, d, false, false);
            d = __builtin_amdgcn_wmma_f32_16x16x4_f32(false, a2, false, b2, (short)0, d, false, false);
            d = __builtin_amdgcn_wmma_f32_16x16x4_f32(false, a3, false, b3, (short)0, d, false, false);

            // D 16x16: VGPR r = output row r + 8*half at this column.
            // rows 0-7 -> new Re[J=r] (lanes 0-15), rows 8-15 -> new Im[J=r].
            float* dstP = half ? dstIm : dstRe;
            #pragma unroll
            for (int r = 0; r < 8; ++r) {
                const unsigned mo = mbase | ((unsigned)r << lb);
                dstP[mo] = d[r];
            }
        }
        __syncthreads();
        float* tR = srcRe; srcRe = dstRe; dstRe = tR;
        float* tI = srcIm; srcIm = dstIm; dstIm = tI;
    }
    // after 4 stages state is back in reA/imA (== srcRe/srcIm)

    // ---- CNOT ring folded into gather indices + observable quadratic form ----
    float acc = 0.f;
    #pragma unroll
    for (int r = tid; r < 1024; r += THREADS) {
        float vr[4], vi[4];
        #pragma unroll
        for (int i = 0; i < 4; ++i) {
            unsigned idx = ((unsigned)i << 10) | (unsigned)r;
            #pragma unroll
            for (int q = NQ-1; q >= 0; --q) {       // psi_f[m] = psi_rot[p0(p1(...p11(m)))]
                const int cpos = NQ-1 - q;
                const int tpos = NQ-1 - ((q+1) % NQ);
                idx ^= ((idx >> cpos) & 1u) << tpos;
            }
            vr[i] = srcRe[idx];
            vi[i] = srcIm[idx];
        }
        #pragma unroll
        for (int i = 0; i < 4; ++i)
            #pragma unroll
            for (int j = 0; j < 4; ++j)
                acc += Osh[i*4+j] * (vr[i]*vr[j] + vi[i]*vi[j]);
    }
    __syncthreads();
    red[tid] = acc;
    __syncthreads();
    for (int off = THREADS/2; off > 0; off >>= 1) {
        if (tid < off) red[tid] += red[tid + off];
        __syncthreads();
    }
    if (tid == 0) out[b] = red[0];
}

extern "C" void kernel_launch(void* const* d_in, const int* in_sizes, int n_in,
                              void* d_out, int out_size, void* d_ws, size_t ws_size,
                              hipStream_t stream)
{
    const float* inputs     = (const float*)d_in[0];   // (B, 4096) f32
    const float* observable = (const float*)d_in[1];   // (4, 4)    f32
    const float* w          = (const float*)d_in[2];   // (1,12,3)  f32
    float* out = (float*)d_out;                        // (B,)      f32
    const int B = in_sizes[0] / DIM;
    amp_model_kernel<<<B, THREADS, 0, stream>>>(inputs, observable, w, out);
}